// ViT_15470472200194
// MI455X (gfx1250) — compile-verified
//
#include <hip/hip_runtime.h>

// ---------------------------------------------------------------------------
// ViT head, collapsed: reference only consumes tokens.reshape(-1)[:10]
//   = softmax( W[0:10,:] @ patch00(image0) )
// patch00[d] with d = c*256 + i*16 + j  <->  images[0, c, i, j], i,j in [0,16)
// Single wave32 workgroup; exact-f32 path via V_WMMA_F32_16X16X4_F32.
// ---------------------------------------------------------------------------

typedef __attribute__((ext_vector_type(2))) float v2f;
typedef __attribute__((ext_vector_type(8))) float v8f;

__device__ __forceinline__ v8f wmma_f32_k4(const float* __restrict__ wrow,
                                           const float* __restrict__ pp,
                                           int kb, v8f c)
{
    v2f a; a.x = wrow[kb]; a.y = wrow[kb + 1];
    v2f b; b.x = pp[kb];   b.y = pp[kb + 1];
    // 8 args: (neg_a, A, neg_b, B, c_mod, C, reuse_a, reuse_b)
    return __builtin_amdgcn_wmma_f32_16x16x4_f32(false, a, false, b,
                                                 (short)0, c, false, false);
}

__launch_bounds__(32)
__global__ void vit_head_wmma(const float* __restrict__ img,
                              const float* __restrict__ W,
                              float* __restrict__ out)
{
    __shared__ float p[768];   // patch (0,0) of image 0, flattened (c, i, j)
    __shared__ float acc[16];  // D-matrix column 0 (out[0..15])

    const int lane = threadIdx.x;  // 0..31, one wave

    // ---- Stage patch into LDS: d = c*256 + i*16 + j -> img[c*50176 + i*224 + j]
    for (int t = lane; t < 768; t += 32) {
        const int c = t >> 8;
        const int r = t & 255;
        const int i = r >> 4;
        const int j = r & 15;
        p[t] = img[c * 50176 + i * 224 + j];
    }
    __syncthreads();

    // ---- WMMA GEMV: out[m] = sum_k W[m][k] * p[k],  m = 0..15 (10 used)
    const int m    = lane & 15;
    const int half = lane >> 4;                 // K sub-lane group
    const float* wrow = W + m * 768 + 2 * half; // A: lane holds M=m, K=r+2*half
    const float* pp   = p + 2 * half;           // B: broadcast across N

    __builtin_prefetch(wrow, 0, 1);             // global_prefetch_b8

    v8f c0 = {}, c1 = {}, c2 = {}, c3 = {};     // 4 accumulators: break RAW chain
    #pragma unroll 2
    for (int kb = 0; kb < 768; kb += 16) {
        c0 = wmma_f32_k4(wrow, pp, kb + 0,  c0);
        c1 = wmma_f32_k4(wrow, pp, kb + 4,  c1);
        c2 = wmma_f32_k4(wrow, pp, kb + 8,  c2);
        c3 = wmma_f32_k4(wrow, pp, kb + 12, c3);
    }
    #pragma unroll
    for (int r = 0; r < 8; ++r)
        c0[r] = (c0[r] + c1[r]) + (c2[r] + c3[r]);

    // D layout: lane L, VGPR r -> M = r + 8*(L>>4), N = L&15.
    // B was broadcast over N, so every lane's c0[r] == out[r + 8*half].
    if (m == 0) {
        #pragma unroll
        for (int r = 0; r < 8; ++r) acc[r + 8 * half] = c0[r];
    }
    __syncthreads();

    // ---- Softmax over the 10 live logits, single lane (trivially cheap)
    if (lane == 0) {
        float mx = acc[0];
        #pragma unroll
        for (int k = 1; k < 10; ++k) mx = fmaxf(mx, acc[k]);
        float e[10];
        float s = 0.0f;
        #pragma unroll
        for (int k = 0; k < 10; ++k) { e[k] = expf(acc[k] - mx); s += e[k]; }
        const float inv = 1.0f / s;
        #pragma unroll
        for (int k = 0; k < 10; ++k) out[k] = e[k] * inv;
    }
}

extern "C" void kernel_launch(void* const* d_in, const int* in_sizes, int n_in,
                              void* d_out, int out_size, void* d_ws, size_t ws_size,
                              hipStream_t stream)
{
    const float* img = (const float*)d_in[0];  // (256, 3, 224, 224) f32
    const float* W   = (const float*)d_in[1];  // (768, 768) f32
    float* out       = (float*)d_out;          // (1, 10) f32

    vit_head_wmma<<<1, 32, 0, stream>>>(img, W, out);
}